// CollapsedPBFAOptimized_54589034332914
// MI455X (gfx1250) — compile-verified
//
#include <hip/hip_runtime.h>
#include <hip/hip_bf16.h>

#define D_MODEL 1024
#define N_HEADS 16
#define D_HEAD  64
#define CHEB_M  11      // 2*ORDER-1
#define SEQ     4096
#define BATCH   2
#define FDIM    (CHEB_M * D_HEAD)   // 704 flattened (m,d) feature dim

typedef __attribute__((ext_vector_type(2))) float v2f;
typedef __attribute__((ext_vector_type(8))) float v8f;

// D(16x16,f32) = A(16x4,f32) * B(4x16,f32) + C  -- fp32 matrix pipe
__device__ __forceinline__ v8f wmma4(v2f a, v2f b, v8f c) {
  return __builtin_amdgcn_wmma_f32_16x16x4_f32(false, a, false, b,
                                               (short)0, c, false, false);
}

// Async 16B/lane copy global -> LDS (VGLOBAL GLOBAL_LOAD_ASYNC_TO_LDS_B128,
// tracked by ASYNCcnt; per-lane LDS byte offset + per-lane 64-bit address).
// Generic shared-pointer low 32 bits are the LDS offset (aperture layout).
__device__ __forceinline__ void async_g2l_b128(void* lds_dst, const void* gsrc) {
  uint32_t loff = (uint32_t)(uintptr_t)lds_dst;
  asm volatile("global_load_async_to_lds_b128 %0, %1, off"
               :: "v"(loff), "v"(gsrc) : "memory");
}
template <int N>
__device__ __forceinline__ void wait_asynccnt() {
  asm volatile("s_wait_asynccnt %0" :: "n"(N) : "memory");
}

// Chebyshev T_m(x), m in [0,10]; branchless select so EXEC stays all-ones.
__device__ __forceinline__ float chebT(float x, int m) {
  float tp = 1.0f, tc = x;
  float r = (m == 0) ? 1.0f : x;
#pragma unroll
  for (int i = 2; i < CHEB_M; ++i) {
    float tn = __builtin_fmaf(2.0f * x, tc, -tp);
    tp = tc; tc = tn;
    r = (i == m) ? tn : r;
  }
  return r;
}

__device__ __forceinline__ float sclamp(float v) {  // scale by Dh^-0.5 then clamp
  return fminf(fmaxf(v * 0.125f, -1.0f), 1.0f);
}

// ---------------------------------------------------------------------------
// GEMM (NT): C[M][N] = A[M][K] * B[N][K]^T.  Block tile 128x64, K-chunk 32,
// async double-buffered LDS staging (one barrier per chunk).
// 8 waves; wave w owns M-strip [16w,16w+16) x full 64 N (4 accumulators).
// ---------------------------------------------------------------------------
#define GPAD 36   // %4==0 (16B-aligned B128), 36r mod 64 distinct for r=0..15
__global__ __launch_bounds__(256)
void gemm_nt_wmma(float* __restrict__ C, const float* __restrict__ A,
                  const float* __restrict__ B, int N, int K) {
  __shared__ __align__(16) float As[2][128 * GPAD];
  __shared__ __align__(16) float Bs[2][64 * GPAD];
  const int tid  = threadIdx.x;
  const int wave = tid >> 5, lane = tid & 31;
  const int lm   = lane & 15, lh = lane >> 4;
  const int tileM = blockIdx.x * 128;
  const int tileN = blockIdx.y * 64;

  auto stage = [&](int kc, int buf) {           // 6 async B128 ops / thread
#pragma unroll
    for (int i = 0; i < 4; ++i) {               // A panel: 128x32
      int idx = i * 256 + tid;
      int r = idx >> 3, c4 = (idx & 7) << 2;
      async_g2l_b128(&As[buf][r * GPAD + c4],
                     A + (size_t)(tileM + r) * K + kc + c4);
    }
#pragma unroll
    for (int i = 0; i < 2; ++i) {               // B panel: 64x32
      int idx = i * 256 + tid;
      int r = idx >> 3, c4 = (idx & 7) << 2;
      async_g2l_b128(&Bs[buf][r * GPAD + c4],
                     B + (size_t)(tileN + r) * K + kc + c4);
    }
  };

  v8f acc[4] = {};
  stage(0, 0);
  int buf = 0;
  for (int kc = 0; kc < K; kc += 32, buf ^= 1) {
    wait_asynccnt<0>();        // own chunk-kc loads done
    __syncthreads();           // => everyone's chunk-kc loads done
    if (kc + 32 < K) stage(kc + 32, buf ^ 1);   // overlap with compute
#pragma unroll
    for (int k0 = 0; k0 < 32; k0 += 4) {
      const float* ap = &As[buf][(wave * 16 + lm) * GPAD + k0 + 2 * lh];
      v2f a; a.x = ap[0]; a.y = ap[1];
#pragma unroll
      for (int nt = 0; nt < 4; ++nt) {
        const float* bp = &Bs[buf][(nt * 16 + lm) * GPAD + k0 + 2 * lh];
        v2f bf; bf.x = bp[0]; bf.y = bp[1];
        acc[nt] = wmma4(a, bf, acc[nt]);
      }
    }
  }
#pragma unroll
  for (int nt = 0; nt < 4; ++nt)
#pragma unroll
    for (int r = 0; r < 8; ++r) {
      int row = tileM + wave * 16 + r + 8 * lh;   // D: VGPR r -> M=r / r+8
      C[(size_t)row * N + tileN + nt * 16 + lm] = acc[nt][r];
    }
}

// ---------------------------------------------------------------------------
// KV state: kv[bh][cm*64+d][v] = sum_s T_cm(khat[s][d]) * v[s][v]
// grid (B*H, 11).  64x64 tile per block for degree cm; K = S = 4096 streamed
// in async double-buffered chunks of 32 rows.
// 8 waves: wave w -> mt = w>>1 (d-tile), nt in {(w&1)*2, (w&1)*2+1}.
// ---------------------------------------------------------------------------
#define KPAD 68
__global__ __launch_bounds__(256)
void cheb_kv_state(float* __restrict__ kvbuf, const float* __restrict__ qkv) {
  __shared__ __align__(16) float Ks[2][32 * KPAD];   // raw k chunk
  __shared__ __align__(16) float Vs[2][32 * KPAD];
  const int tid  = threadIdx.x;
  const int wave = tid >> 5, lane = tid & 31;
  const int lm   = lane & 15, lh = lane >> 4;
  const int bh = blockIdx.x, bat = bh >> 4, h = bh & 15;
  const int cm = blockIdx.y;                      // Chebyshev degree (uniform)
  const int mt = wave >> 1, nt0 = (wave & 1) * 2;
  const size_t rowbase = (size_t)bat * SEQ;

  auto stage = [&](int sc, int buf) {             // 4 async B128 ops / thread
#pragma unroll
    for (int i = 0; i < 2; ++i) {                 // 32 rows x 64 cols, k & v
      int idx = i * 256 + tid;
      int r = idx >> 4, c4 = (idx & 15) << 2;
      const float* src =
          qkv + (rowbase + sc + r) * (3 * D_MODEL) + h * D_HEAD + c4;
      async_g2l_b128(&Ks[buf][r * KPAD + c4], src + D_MODEL);
      async_g2l_b128(&Vs[buf][r * KPAD + c4], src + 2 * D_MODEL);
    }
  };

  v8f acc0 = {}, acc1 = {};
  stage(0, 0);
  int buf = 0;
  for (int sc = 0; sc < SEQ; sc += 32, buf ^= 1) {
    wait_asynccnt<0>();
    __syncthreads();
    if (sc + 32 < SEQ) stage(sc + 32, buf ^ 1);
#pragma unroll
    for (int k0 = 0; k0 < 32; k0 += 4) {
      int s0 = k0 + 2 * lh;
      int d  = mt * 16 + lm;
      v2f a;                                      // clamp+scale at read time
      a.x = chebT(sclamp(Ks[buf][s0 * KPAD + d]), cm);
      a.y = chebT(sclamp(Ks[buf][(s0 + 1) * KPAD + d]), cm);
      v2f b0, b1;
      b0.x = Vs[buf][s0 * KPAD + nt0 * 16 + lm];
      b0.y = Vs[buf][(s0 + 1) * KPAD + nt0 * 16 + lm];
      b1.x = Vs[buf][s0 * KPAD + (nt0 + 1) * 16 + lm];
      b1.y = Vs[buf][(s0 + 1) * KPAD + (nt0 + 1) * 16 + lm];
      acc0 = wmma4(a, b0, acc0);
      acc1 = wmma4(a, b1, acc1);
    }
  }
  float* out = kvbuf + (size_t)bh * (FDIM * D_HEAD);
#pragma unroll
  for (int r = 0; r < 8; ++r) {
    int mrow = cm * 64 + mt * 16 + r + 8 * lh;
    out[(size_t)mrow * D_HEAD + nt0 * 16 + lm]       = acc0[r];
    out[(size_t)mrow * D_HEAD + (nt0 + 1) * 16 + lm] = acc1[r];
  }
}

// ---------------------------------------------------------------------------
// Attention out: attn[b][s][h*64+v] = sum_{cm,d} beta[h][cm]*T_cm(qhat[s][d])
//                                      * kv[bh][cm*64+d][v]
// grid (B*H, S/128).  Wave w -> 16-row s-strip; K = 704; cm uniform per 64
// k-steps so beta is an SGPR broadcast.  kv read once per block straight from
// L2 (exactly 180 KB/block, L2-resident); q tile staged async once.
// ---------------------------------------------------------------------------
__global__ __launch_bounds__(256)
void cheb_attn_out(float* __restrict__ attn, const float* __restrict__ qkv,
                   const float* __restrict__ kvbuf,
                   const float* __restrict__ beta) {
  __shared__ __align__(16) float Qs[128 * KPAD];  // raw q tile
  const int tid  = threadIdx.x;
  const int wave = tid >> 5, lane = tid & 31;
  const int lm   = lane & 15, lh = lane >> 4;
  const int bh = blockIdx.x, bat = bh >> 4, h = bh & 15;
  const int stile = blockIdx.y * 128;

#pragma unroll
  for (int i = 0; i < 8; ++i) {                   // 128 x 64 q tile, async
    int idx = i * 256 + tid;
    int r = idx >> 4, c4 = (idx & 15) << 2;
    const float* src = qkv + ((size_t)bat * SEQ + stile + r) * (3 * D_MODEL)
                       + h * D_HEAD + c4;
    async_g2l_b128(&Qs[r * KPAD + c4], src);
  }
  wait_asynccnt<0>();
  __syncthreads();

  v8f acc[4] = {};
  const float* kvb = kvbuf + (size_t)bh * (FDIM * D_HEAD);
  const int srow = wave * 16 + lm;                // A-frag row (s, local)
  for (int cm = 0; cm < CHEB_M; ++cm) {
    const float bet = beta[h * CHEB_M + cm];      // uniform -> SGPR
    const float* kvc = kvb + (size_t)cm * 64 * D_HEAD;
#pragma unroll 4
    for (int kd = 0; kd < 64; kd += 4) {
      int d0 = kd + 2 * lh;
      v2f a;
      a.x = bet * chebT(sclamp(Qs[srow * KPAD + d0]), cm);
      a.y = bet * chebT(sclamp(Qs[srow * KPAD + d0 + 1]), cm);
#pragma unroll
      for (int nt = 0; nt < 4; ++nt) {
        v2f bf;
        bf.x = kvc[(size_t)d0 * D_HEAD + nt * 16 + lm];
        bf.y = kvc[(size_t)(d0 + 1) * D_HEAD + nt * 16 + lm];
        acc[nt] = wmma4(a, bf, acc[nt]);
      }
    }
  }
#pragma unroll
  for (int nt = 0; nt < 4; ++nt)
#pragma unroll
    for (int r = 0; r < 8; ++r) {
      int so = stile + wave * 16 + r + 8 * lh;
      attn[((size_t)bat * SEQ + so) * D_MODEL + h * D_HEAD + nt * 16 + lm]
          = acc[nt][r];
    }
}

// ---------------------------------------------------------------------------
extern "C" void kernel_launch(void* const* d_in, const int* in_sizes, int n_in,
                              void* d_out, int out_size, void* d_ws, size_t ws_size,
                              hipStream_t stream) {
  const float* x     = (const float*)d_in[0];   // (2,4096,1024)
  const float* w_in  = (const float*)d_in[1];   // (3072,1024)
  const float* w_out = (const float*)d_in[2];   // (1024,1024)
  const float* beta  = (const float*)d_in[3];   // (16,11)
  float* out = (float*)d_out;                   // (2,4096,1024)

  float* qkv   = (float*)d_ws;                          // 8192*3072 f32 (96 MB)
  float* kvbuf = qkv + (size_t)8192 * 3072;             // 32*704*64  (5.8 MB)
  float* attn  = kvbuf + (size_t)32 * FDIM * D_HEAD;    // 8192*1024  (32 MB)

  // 1) qkv = x @ w_in^T        (M=8192, N=3072, K=1024)
  gemm_nt_wmma<<<dim3(64, 48), 256, 0, stream>>>(qkv, x, w_in, 3 * D_MODEL, D_MODEL);
  // 2) kv state per (b,h,cm)   (M=704 via 11x64, N=64, K=4096)
  cheb_kv_state<<<dim3(BATCH * N_HEADS, CHEB_M), 256, 0, stream>>>(kvbuf, qkv);
  // 3) attn = tq @ kv per (b,h) (M=4096, N=64, K=704), written in (B,S,D) layout
  cheb_attn_out<<<dim3(BATCH * N_HEADS, SEQ / 128), 256, 0, stream>>>(attn, qkv, kvbuf, beta);
  // 4) out = attn @ w_out^T    (M=8192, N=1024, K=1024)
  gemm_nt_wmma<<<dim3(64, 16), 256, 0, stream>>>(out, attn, w_out, D_MODEL, D_MODEL);
}